// QuantumLayer_41154376630450
// MI455X (gfx1250) — compile-verified
//
#include <hip/hip_runtime.h>
#include <math.h>

typedef float v2f __attribute__((ext_vector_type(2)));
typedef float v4f __attribute__((ext_vector_type(4)));
typedef float v8f __attribute__((ext_vector_type(8)));

#define DIM 4096
#define THREADS 128
#define CHUNKS 8   // 8 v4f chunks per thread (8*128*4 = 4096)

__device__ __forceinline__ v2f mk2(float a, float b) { v2f r; r.x = a; r.y = b; return r; }

// D = A(16x4) * B(4x16) + C, fp32. Chained over K for K=16.
__device__ __forceinline__ v8f wmma_k4(v2f a, v2f b, v8f c) {
  return __builtin_amdgcn_wmma_f32_16x16x4_f32(false, a, false, b, (short)0, c, false, false);
}

__global__ __launch_bounds__(THREADS) void quantum_circuit_kernel(
    const float* __restrict__ x, const float* __restrict__ params,
    float* __restrict__ out) {
  __shared__ __align__(16) float s_re[DIM];
  __shared__ __align__(16) float s_im[DIM];
  __shared__ float u_re[3][256];   // 16x16 block unitaries (row-major [i][j], amp' = U amp)
  __shared__ float u_im[3][256];
  __shared__ float red[2 * THREADS];
  __shared__ float scal[2];

  const int tid = threadIdx.x;
  const long long base = (long long)blockIdx.x * DIM;
  v4f* s_re4 = (v4f*)s_re;
  v4f* s_im4 = (v4f*)s_im;

  // ---------- Stage row into LDS via CDNA5 async global->LDS DMA ----------
  {
    const float* gsrc = x + base;                       // wave-uniform 64-bit base (SGPR pair)
    const unsigned ldsbase = (unsigned)(uintptr_t)(void*)s_re; // low 32 bits = LDS offset
#pragma unroll
    for (int i = 0; i < CHUNKS; ++i) {
      unsigned c = tid + i * THREADS;
      unsigned voff = c * 16u;                          // per-lane byte offset
      unsigned ldsa = ldsbase + voff;                   // per-lane LDS byte address
      asm volatile("global_load_async_to_lds_b128 %0, %1, %2"
                   :
                   : "v"(ldsa), "v"(voff), "s"(gsrc)
                   : "memory");
    }
    asm volatile("s_wait_asynccnt 0x0" ::: "memory");
  }
  __syncthreads();

  // ---------- Phase A: sum(x^2) and nonzero count (from LDS) ----------
  float s2 = 0.f, cnt = 0.f;
#pragma unroll
  for (int i = 0; i < CHUNKS; ++i) {
    v4f v = s_re4[tid + i * THREADS];
    s2 += v.x * v.x + v.y * v.y + v.z * v.z + v.w * v.w;
    cnt += ((v.x != 0.f) ? 1.f : 0.f) + ((v.y != 0.f) ? 1.f : 0.f) +
           ((v.z != 0.f) ? 1.f : 0.f) + ((v.w != 0.f) ? 1.f : 0.f);
  }
  red[tid] = s2;
  red[THREADS + tid] = cnt;
  __syncthreads();
  if (tid == 0) {
    float a = 0.f, b = 0.f;
    for (int i = 0; i < THREADS; ++i) { a += red[i]; b += red[THREADS + i]; }
    scal[0] = a; scal[1] = b;
  }
  __syncthreads();
  const float rn = 1.0f / (sqrtf(scal[0]) + 1e-8f);
  const float invq = rsqrtf(scal[1]); // sqrt(#nonzero); matches reference (NaN if all-zero)

  // ---------- Phase B: amplitude encoding in place ----------
#pragma unroll
  for (int i = 0; i < CHUNKS; ++i) {
    int c = tid + i * THREADS;
    v4f xv = s_re4[c];
    v4f r4, i4;
#pragma unroll
    for (int k = 0; k < 4; ++k) {
      float v = xv[k] * rn;
      float mag = fminf(fabsf(v), 1.0f);
      float sg = (v > 0.f) ? 1.f : ((v < 0.f) ? -1.f : 0.f);
      r4[k] = sg * mag * invq;
      i4[k] = sg * sqrtf(fmaxf(1.f - mag * mag, 0.f)) * invq;
    }
    s_re4[c] = r4;
    s_im4[c] = i4;
  }

  // ---------- Build the three 4-qubit block unitaries ----------
  // Block s covers qubits 4s..4s+3: H(0),[CRX(0,1),CNOT(0,1)],H(1),...,H(3)
  // Thread (s,j) propagates basis column j; U[:,j] = circuit(e_j).
  if (tid < 48) {
    int s = tid >> 4, j = tid & 15;
    float* ur = &u_re[s][0];
    float* ui = &u_im[s][0];
    for (int i = 0; i < 16; ++i) { ur[i * 16 + j] = (i == j) ? 1.f : 0.f; ui[i * 16 + j] = 0.f; }
    const float R = 0.70710678118654752f;
    for (int lq = 0; lq < 4; ++lq) {
      int st = 1 << lq;
      for (int i = 0; i < 16; ++i) {            // H on local qubit lq
        if (i & st) continue;
        float ar = ur[i * 16 + j], ai = ui[i * 16 + j];
        float br = ur[(i + st) * 16 + j], bi = ui[(i + st) * 16 + j];
        ur[i * 16 + j] = (ar + br) * R;        ui[i * 16 + j] = (ai + bi) * R;
        ur[(i + st) * 16 + j] = (ar - br) * R; ui[(i + st) * 16 + j] = (ai - bi) * R;
      }
      if (lq < 3) {                              // CRX(lq->lq+1, params[4s+lq]) then CNOT(lq,lq+1)
        int tt = st << 1;
        float th = params[4 * s + lq] * 0.5f;
        float c = __cosf(th), sn = __sinf(th);
        for (int i = 0; i < 16; ++i) {
          if (!(i & st) || (i & tt)) continue;   // control=1, target=0 slot
          float ar = ur[i * 16 + j], ai = ui[i * 16 + j];
          float br = ur[(i + tt) * 16 + j], bi = ui[(i + tt) * 16 + j];
          float nar = c * ar + sn * bi, nai = c * ai - sn * br;  // a' = c*a - i*s*b
          float nbr = sn * ai + c * br, nbi = -sn * ar + c * bi; // b' = -i*s*a + c*b
          ur[i * 16 + j] = nbr;        ui[i * 16 + j] = nbi;     // CNOT swaps a'<->b'
          ur[(i + tt) * 16 + j] = nar; ui[(i + tt) * 16 + j] = nai;
        }
      }
    }
  }
  __syncthreads();

  const int lane = tid & 31;
  const int wid  = (tid >> 5) & 3;   // 4 waves, provable range for unrolling
  const int m16  = lane & 15;
  const int hh   = lane >> 4;        // lane half selects K/M upper groups per ISA layouts

  // ---------- Stage 1: B0 on bits 0-3.  S(256x16) <- S * B0^T ----------
  {
    v2f br_[4], bi_[4], nbi_[4];
#pragma unroll
    for (int kc = 0; kc < 4; ++kc) {
      int k0 = kc * 4 + hh * 2;                 // B[k][n] = B0[n][k]
      br_[kc]  = mk2(u_re[0][m16 * 16 + k0], u_re[0][m16 * 16 + k0 + 1]);
      bi_[kc]  = mk2(u_im[0][m16 * 16 + k0], u_im[0][m16 * 16 + k0 + 1]);
      nbi_[kc] = -bi_[kc];
    }
#pragma unroll 4
    for (int rt = wid; rt < 16; rt += 4) {      // 16 row-tiles of 16 rows
      v2f ar_[4], ai_[4];
#pragma unroll
      for (int kc = 0; kc < 4; ++kc) {
        int k0 = kc * 4 + hh * 2;
        int rb = (rt * 16 + m16) * 16;          // A[m][k] = S[row m][k]
        ar_[kc] = mk2(s_re[rb + k0], s_re[rb + k0 + 1]);
        ai_[kc] = mk2(s_im[rb + k0], s_im[rb + k0 + 1]);
      }
      v8f accr = {0.f,0.f,0.f,0.f,0.f,0.f,0.f,0.f};
      v8f acci = {0.f,0.f,0.f,0.f,0.f,0.f,0.f,0.f};
#pragma unroll
      for (int kc = 0; kc < 4; ++kc) {
        accr = wmma_k4(ar_[kc], br_[kc], accr);
        accr = wmma_k4(ai_[kc], nbi_[kc], accr);
        acci = wmma_k4(ar_[kc], bi_[kc], acci);
        acci = wmma_k4(ai_[kc], br_[kc], acci);
      }
#pragma unroll
      for (int v = 0; v < 8; ++v) {
        int idx = (rt * 16 + hh * 8 + v) * 16 + m16;
        s_re[idx] = accr[v]; s_im[idx] = acci[v];
      }
    }
  }
  __syncthreads();

  // ---------- G(3,4): CRX(3->4,params[3]) then CNOT(3,4): only l=1,l=3 change ----------
  {
    float th = params[3] * 0.5f;
    float c = __cosf(th), sn = __sinf(th);
#pragma unroll
    for (int g = tid; g < 1024; g += THREADS) {
      int i0 = (g & 7) | ((g >> 3) << 5);       // bits 3,4 = 0
      int i1 = i0 + 8, i3 = i0 + 24;
      float a1r = s_re[i1], a1i = s_im[i1], a3r = s_re[i3], a3i = s_im[i3];
      s_re[i1] = sn * a1i + c * a3r;  s_im[i1] = -sn * a1r + c * a3i;
      s_re[i3] = c * a1r + sn * a3i;  s_im[i3] = c * a1i - sn * a3r;
    }
  }
  __syncthreads();

  // ---------- Stage 3: B1 on bits 4-7.  For each a: M_a(16x16) <- B1 * M_a ----------
  {
    v2f uar[4], uai[4], nuai[4];
#pragma unroll
    for (int kc = 0; kc < 4; ++kc) {
      int k0 = kc * 4 + hh * 2;                 // A[m][k] = U1[m][k]
      uar[kc]  = mk2(u_re[1][m16 * 16 + k0], u_re[1][m16 * 16 + k0 + 1]);
      uai[kc]  = mk2(u_im[1][m16 * 16 + k0], u_im[1][m16 * 16 + k0 + 1]);
      nuai[kc] = -uai[kc];
    }
#pragma unroll 4
    for (int a = wid; a < 16; a += 4) {
      v2f brr[4], bii[4];
#pragma unroll
      for (int kc = 0; kc < 4; ++kc) {
        int k0 = kc * 4 + hh * 2;
        int b0 = a * 256 + k0 * 16 + m16;       // B[k][n] = amp[a, k, n]
        brr[kc] = mk2(s_re[b0], s_re[b0 + 16]);
        bii[kc] = mk2(s_im[b0], s_im[b0 + 16]);
      }
      v8f accr = {0.f,0.f,0.f,0.f,0.f,0.f,0.f,0.f};
      v8f acci = {0.f,0.f,0.f,0.f,0.f,0.f,0.f,0.f};
#pragma unroll
      for (int kc = 0; kc < 4; ++kc) {
        accr = wmma_k4(uar[kc], brr[kc], accr);
        accr = wmma_k4(nuai[kc], bii[kc], accr);
        acci = wmma_k4(uar[kc], bii[kc], acci);
        acci = wmma_k4(uai[kc], brr[kc], acci);
      }
#pragma unroll
      for (int v = 0; v < 8; ++v) {
        int idx = a * 256 + (hh * 8 + v) * 16 + m16;
        s_re[idx] = accr[v]; s_im[idx] = acci[v];
      }
    }
  }
  __syncthreads();

  // ---------- G(7,8) ----------
  {
    float th = params[7] * 0.5f;
    float c = __cosf(th), sn = __sinf(th);
#pragma unroll
    for (int g = tid; g < 1024; g += THREADS) {
      int i0 = (g & 127) | ((g >> 7) << 9);     // bits 7,8 = 0
      int i1 = i0 + 128, i3 = i0 + 384;
      float a1r = s_re[i1], a1i = s_im[i1], a3r = s_re[i3], a3i = s_im[i3];
      s_re[i1] = sn * a1i + c * a3r;  s_im[i1] = -sn * a1r + c * a3i;
      s_re[i3] = c * a1r + sn * a3i;  s_im[i3] = c * a1i - sn * a3r;
    }
  }
  __syncthreads();

  // ---------- Stage 5: B2 on bits 8-11.  M(16x256) <- B2 * M ----------
  {
    v2f uar[4], uai[4], nuai[4];
#pragma unroll
    for (int kc = 0; kc < 4; ++kc) {
      int k0 = kc * 4 + hh * 2;
      uar[kc]  = mk2(u_re[2][m16 * 16 + k0], u_re[2][m16 * 16 + k0 + 1]);
      uai[kc]  = mk2(u_im[2][m16 * 16 + k0], u_im[2][m16 * 16 + k0 + 1]);
      nuai[kc] = -uai[kc];
    }
#pragma unroll 4
    for (int t = wid; t < 16; t += 4) {         // 16 column-tiles (b = t, all c)
      v2f brr[4], bii[4];
#pragma unroll
      for (int kc = 0; kc < 4; ++kc) {
        int k0 = kc * 4 + hh * 2;
        int b0 = k0 * 256 + t * 16 + m16;       // B[k][n] = amp[k, bc = t*16+n]
        brr[kc] = mk2(s_re[b0], s_re[b0 + 256]);
        bii[kc] = mk2(s_im[b0], s_im[b0 + 256]);
      }
      v8f accr = {0.f,0.f,0.f,0.f,0.f,0.f,0.f,0.f};
      v8f acci = {0.f,0.f,0.f,0.f,0.f,0.f,0.f,0.f};
#pragma unroll
      for (int kc = 0; kc < 4; ++kc) {
        accr = wmma_k4(uar[kc], brr[kc], accr);
        accr = wmma_k4(nuai[kc], bii[kc], accr);
        acci = wmma_k4(uar[kc], bii[kc], acci);
        acci = wmma_k4(uai[kc], brr[kc], acci);
      }
#pragma unroll
      for (int v = 0; v < 8; ++v) {
        int idx = (hh * 8 + v) * 256 + t * 16 + m16;
        s_re[idx] = accr[v]; s_im[idx] = acci[v];
      }
    }
  }
  __syncthreads();

  // ---------- CRX(11->0, params[11]): control bit11, target bit0 ----------
  {
    float th = params[11] * 0.5f;
    float c = __cosf(th), sn = __sinf(th);
#pragma unroll
    for (int g = tid; g < 1024; g += THREADS) {
      int idx = 2048 + (g << 1);                // bit11=1, bit0=0
      float ar = s_re[idx], ai = s_im[idx];
      float br = s_re[idx + 1], bi = s_im[idx + 1];
      s_re[idx]     = c * ar + sn * bi;   s_im[idx]     = c * ai - sn * br;
      s_re[idx + 1] = sn * ai + c * br;   s_im[idx + 1] = -sn * ar + c * bi;
    }
  }
  __syncthreads();

  // ---------- Probabilities (vectorized, non-temporal stores) ----------
  {
    v4f* out4 = (v4f*)out;
    long long obase = (long long)blockIdx.x * (DIM / 4);
#pragma unroll
    for (int i = 0; i < CHUNKS; ++i) {
      int c = tid + i * THREADS;
      v4f r = s_re4[c];
      v4f m = s_im4[c];
      v4f o = r * r + m * m;
      __builtin_nontemporal_store(o, &out4[obase + c]);
    }
  }
}

extern "C" void kernel_launch(void* const* d_in, const int* in_sizes, int n_in,
                              void* d_out, int out_size, void* d_ws, size_t ws_size,
                              hipStream_t stream) {
  const float* x      = (const float*)d_in[0];  // (8,512,4096) fp32
  const float* params = (const float*)d_in[1];  // (12,) fp32
  float* out = (float*)d_out;                   // (8,512,4096) fp32
  int nstates = in_sizes[0] / DIM;              // 4096
  quantum_circuit_kernel<<<nstates, THREADS, 0, stream>>>(x, params, out);
}